// LINK_16604343566783
// MI455X (gfx1250) — compile-verified
//
#include <hip/hip_runtime.h>
#include <climits>
#include <cstdint>

#define OUTC 32
#define SCATTER_BLOCK 256
#define WAVES_PER_BLOCK (SCATTER_BLOCK / 32)

// ---------------------------------------------------------------------------
// Kernel 1: out[n][c] = b[c] for all nodes; ws_min = INT_MAX.
// ---------------------------------------------------------------------------
__global__ void link_init_kernel(float* __restrict__ out,
                                 const float* __restrict__ b,
                                 int total, int* __restrict__ wsMin) {
    int tid = blockIdx.x * blockDim.x + threadIdx.x;
    if (tid == 0) *wsMin = INT_MAX;
    int stride = gridDim.x * blockDim.x;
    for (int i = tid; i < total; i += stride)
        out[i] = b[i & (OUTC - 1)];
}

// ---------------------------------------------------------------------------
// Kernel 2: global min of row[] (faithful to `row - row.min()`).
// ---------------------------------------------------------------------------
__global__ void link_min_kernel(const int* __restrict__ row, int E,
                                int* __restrict__ wsMin) {
    int tid = blockIdx.x * blockDim.x + threadIdx.x;
    int stride = gridDim.x * blockDim.x;
    int local = INT_MAX;
    for (int i = tid; i < E; i += stride) {
        if (i + stride < E) __builtin_prefetch(&row[i + stride], 0, 0);
        int v = row[i];
        local = (v < local) ? v : local;
    }
#pragma unroll
    for (int off = 16; off > 0; off >>= 1) {
        int o = __shfl_xor(local, off, 32);
        local = (o < local) ? o : local;
    }
    if ((threadIdx.x & 31) == 0)
        atomicMin(wsMin, local);
}

// ---------------------------------------------------------------------------
// Kernel 3: transpose W[32][N] -> Wt[N][32] via LDS tile so per-edge gathers
// become single 128B coalesced loads.
// ---------------------------------------------------------------------------
__global__ __launch_bounds__(1024)
void link_transpose_kernel(const float* __restrict__ W,
                           float* __restrict__ Wt, int N) {
    __shared__ float tile[32][33];
    int n0 = blockIdx.x * 32;
    int tx = threadIdx.x, ty = threadIdx.y;
    if (n0 + tx < N) tile[ty][tx] = W[(size_t)ty * N + (n0 + tx)];
    __syncthreads();
    if (n0 + ty < N) Wt[(size_t)(n0 + ty) * OUTC + tx] = tile[tx][ty];
}

// ---------------------------------------------------------------------------
// Kernel 4: scatter-add with ASYNC double-buffered index staging.
// Each wave owns a private LDS double buffer of 32 (row,col) pairs. The next
// chunk's indices are copied memory->LDS with global_load_async_to_lds_b32
// (ASYNCcnt) while the current chunk's gathers + global_atomic_add_f32
// scatters execute; s_wait_asynccnt 2 retires only the current buffer
// (async loads complete in order). lane == output channel, so both the Wt
// gather and the atomic burst are single 128B wave transactions.
// ---------------------------------------------------------------------------
template <bool TRANSPOSED>
__global__ __launch_bounds__(SCATTER_BLOCK)
void link_scatter_kernel(const int* __restrict__ row,
                         const int* __restrict__ col,
                         const float* __restrict__ Wsrc,
                         float* __restrict__ out,
                         int E, int N,
                         const int* __restrict__ wsMin) {
    __shared__ int sbuf[WAVES_PER_BLOCK][2][64];  // [wave][buf][32 row | 32 col]

    const int lane   = threadIdx.x & 31;
    const int wblk   = threadIdx.x >> 5;
    const int wave   = (blockIdx.x * blockDim.x + threadIdx.x) >> 5;
    const int nwaves = (gridDim.x * blockDim.x) >> 5;
    const int rmin   = *wsMin;
    const int step   = nwaves * 32;

    // Low 32 bits of the generic shared-space address == LDS byte address.
    uint32_t ldsRow[2], ldsCol[2];
    ldsRow[0] = (uint32_t)(uintptr_t)&sbuf[wblk][0][lane];
    ldsCol[0] = (uint32_t)(uintptr_t)&sbuf[wblk][0][32 + lane];
    ldsRow[1] = (uint32_t)(uintptr_t)&sbuf[wblk][1][lane];
    ldsCol[1] = (uint32_t)(uintptr_t)&sbuf[wblk][1][32 + lane];

    int base = wave * 32;
    if (base >= E) return;

    auto stage = [&](int b, int buf) {
        int e = b + lane;
        if (e >= E) e = E - 1;                 // clamp; slot never consumed
        uint32_t off = (uint32_t)e * 4u;
        asm volatile("global_load_async_to_lds_b32 %0, %1, %2"
                     :: "v"(ldsRow[buf]), "v"(off), "s"(row) : "memory");
        asm volatile("global_load_async_to_lds_b32 %0, %1, %2"
                     :: "v"(ldsCol[buf]), "v"(off), "s"(col) : "memory");
    };

    stage(base, 0);                            // prologue: stage chunk 0
    int buf = 0;
    for (; base < E; base += step, buf ^= 1) {
        const int nbase = base + step;
        const bool hasNext = (nbase < E);
        if (hasNext) {
            stage(nbase, buf ^ 1);             // overlap next chunk's fetch
            asm volatile("s_wait_asynccnt 0x2" ::: "memory");  // cur buf ready
        } else {
            asm volatile("s_wait_asynccnt 0x0" ::: "memory");
        }
        int cnt = E - base; if (cnt > 32) cnt = 32;
        const int* cur = &sbuf[wblk][buf][0];
        for (int j = 0; j < cnt; ++j) {
            int r = cur[j] - rmin;             // ds_load_b32 broadcast
            int c = cur[32 + j];
            float v = TRANSPOSED ? Wsrc[(size_t)c * OUTC + lane]   // 128B coalesced
                                 : Wsrc[(size_t)lane * N + c];     // fallback
            atomicAdd(&out[(size_t)r * OUTC + lane], v);           // L2 atomics
        }
    }
}

// ---------------------------------------------------------------------------
// Host-side launcher (graph-capture safe: stream-ordered kernels only).
// d_in[0]: edge_index int32 [2, E]   d_in[1]: W f32 [32, N]   d_in[2]: b f32 [32]
// d_out:   f32 [N, 32]
// ---------------------------------------------------------------------------
extern "C" void kernel_launch(void* const* d_in, const int* in_sizes, int n_in,
                              void* d_out, int out_size, void* d_ws, size_t ws_size,
                              hipStream_t stream) {
    const int*   edge = (const int*)d_in[0];
    const float* W    = (const float*)d_in[1];
    const float* b    = (const float*)d_in[2];
    float* out = (float*)d_out;

    const int E = in_sizes[0] / 2;
    const int N = in_sizes[1] / OUTC;
    const int* row = edge;
    const int* col = edge + E;

    int*   wsMin = (int*)d_ws;
    float* Wt    = (float*)((char*)d_ws + 256);
    const size_t wtBytes = (size_t)N * OUTC * sizeof(float);
    const bool useWt = (ws_size >= 256 + wtBytes);

    link_init_kernel<<<1024, 256, 0, stream>>>(out, b, N * OUTC, wsMin);
    link_min_kernel <<<1024, 256, 0, stream>>>(row, E, wsMin);

    if (useWt) {
        dim3 tb(32, 32, 1);
        link_transpose_kernel<<<(N + 31) / 32, tb, 0, stream>>>(W, Wt, N);
        link_scatter_kernel<true><<<4096, SCATTER_BLOCK, 0, stream>>>(row, col, Wt, out, E, N, wsMin);
    } else {
        link_scatter_kernel<false><<<4096, SCATTER_BLOCK, 0, stream>>>(row, col, W, out, E, N, wsMin);
    }
}